// CrossAgentAttention_8383776162155
// MI455X (gfx1250) — compile-verified
//
#include <hip/hip_runtime.h>
#include <hip/hip_bf16.h>
#include <math.h>
#include <stdint.h>

// ---------------------------------------------------------------------------
// Cross-agent attention for MI455X (gfx1250, wave32).
//   B=1024, N=32 agents, D=512, H=4 heads, HD=128, all fp32.
//
// Primary path (needs 256 MB workspace): tiled 3-kernel pipeline
//   1) QKV = X @ W1^T + b1      (32768x1536, 128x128 WMMA tiles)
//   2) per-(b,h) masked attention (WMMA scores/ctx + row softmax)
//   3) out = ctx @ W2^T + b2    (32768x512)
// GEMM K-slices are now DOUBLE-BUFFERED and filled with CDNA5 async
// LDS loads (GLOBAL_LOAD_ASYNC_TO_LDS_B128, ASYNCcnt): the L2->LDS copy
// overlaps the WMMA burst on the other buffer and skips the VGPR round trip.
//
// Fallback path (no workspace): fully-fused one-WG-per-batch kernel.
// All matrix math: V_WMMA_F32_16X16X4_F32 (bit-faithful fp32; the problem is
// memory-bound so fp32 matrix rate is ample).
// ---------------------------------------------------------------------------

#define BATCH   1024
#define NA      32
#define DMODEL  512
#define D3      1536
#define NHEAD   4
#define HDIM    128
#define MTOT    (BATCH * NA)     // 32768 rows

typedef float v2f __attribute__((ext_vector_type(2)));
typedef float v8f __attribute__((ext_vector_type(8)));

static __device__ __forceinline__ v8f wmma4(v2f a, v2f b, v8f c) {
  return __builtin_amdgcn_wmma_f32_16x16x4_f32(
      false, a, false, b, (short)0, c, false, false);
}

// Async L2->LDS copy, 16 B per lane, tracked by ASYNCcnt (CDNA5 ISA).
static __device__ __forceinline__ void async_ld128(uint32_t ldsAddr,
                                                   uint64_t gAddr) {
  asm volatile("global_load_async_to_lds_b128 %0, %1, off"
               :: "v"(ldsAddr), "v"(gAddr) : "memory");
}

// ===========================================================================
// Kernel 1/3: C(M x N) = A(M x 512) @ W(N x 512)^T + bias, row-major.
// 128x128 block tile; K staged 32 at a time, double-buffered via async-LDS.
// 8 waves in 2x4 grid, each wave owns a 64x32 register tile (8 v8f accums).
// ===========================================================================
#define KST 36   // LDS K-slice stride: 36 floats = 144 B (16B aligned rows)
#define NSLICE (DMODEL / 32)     // 16 K-slices

__global__ void __launch_bounds__(256)
gemm_nt_bias_kernel(const float* __restrict__ A, const float* __restrict__ W,
                    const float* __restrict__ bias, float* __restrict__ C,
                    int N)
{
  __shared__ float sA[2][128 * KST];
  __shared__ float sW[2][128 * KST];

  const int tid  = threadIdx.x;
  const int wave = tid >> 5, lane = tid & 31;
  const int half = lane >> 4, lm = lane & 15;
  const int mBase = blockIdx.x * 128;
  const int nBase = blockIdx.y * 128;
  const int row0  = (wave >> 2) * 64;   // wave m-offset (0 or 64)
  const int col0  = (wave & 3) * 32;    // wave n-offset (0..96)

  // Per-thread staging slots: 4 float4 per matrix per slice (128x32 tile).
  uint32_t ldsA[4], ldsW[4];            // LDS byte addresses (buffer 0)
  uint64_t gA[4], gW[4];                // global byte addresses (slice 0)
#pragma unroll
  for (int i = 0; i < 4; ++i) {
    int g = tid + i * 256, r = g >> 3, c4 = (g & 7) << 2;
    ldsA[i] = (uint32_t)(uintptr_t)&sA[0][r * KST + c4];
    ldsW[i] = (uint32_t)(uintptr_t)&sW[0][r * KST + c4];
    gA[i]   = (uint64_t)(uintptr_t)&A[(size_t)(mBase + r) * DMODEL + c4];
    gW[i]   = (uint64_t)(uintptr_t)&W[(size_t)(nBase + r) * DMODEL + c4];
  }
  const uint32_t BUFB = 128 * KST * 4;  // bytes between the two buffers

  v8f acc[4][2] = {};

  // Prologue: fill buffer 0 with slice 0 (8 async ops per thread).
#pragma unroll
  for (int i = 0; i < 4; ++i) {
    async_ld128(ldsA[i], gA[i]);
    async_ld128(ldsW[i], gW[i]);
  }

  for (int s = 0; s < NSLICE; ++s) {
    if (s + 1 < NSLICE) {
      // Kick off next slice into the other buffer, then drain this slice's
      // 8 ops (async loads complete in order -> "<= 8" leaves only the new
      // slice's ops outstanding).
      const uint32_t nbo  = (uint32_t)((s + 1) & 1) * BUFB;
      const uint64_t gofs = (uint64_t)(s + 1) * 128;   // 32 floats
#pragma unroll
      for (int i = 0; i < 4; ++i) {
        async_ld128(ldsA[i] + nbo, gA[i] + gofs);
        async_ld128(ldsW[i] + nbo, gW[i] + gofs);
      }
      asm volatile("s_wait_asynccnt 0x8" ::: "memory");
    } else {
      asm volatile("s_wait_asynccnt 0x0" ::: "memory");
    }
    __syncthreads();                     // slice s resident for all threads

    const float* cA = &sA[s & 1][0];
    const float* cW = &sW[s & 1][0];
#pragma unroll
    for (int kb = 0; kb < 32; kb += 4) {
      const int col = kb + half * 2;
      v2f b0 = *(const v2f*)&cW[(col0 + lm) * KST + col];
      v2f b1 = *(const v2f*)&cW[(col0 + 16 + lm) * KST + col];
#pragma unroll
      for (int ms = 0; ms < 4; ++ms) {
        v2f a = *(const v2f*)&cA[(row0 + ms * 16 + lm) * KST + col];
        acc[ms][0] = wmma4(a, b0, acc[ms][0]);
        acc[ms][1] = wmma4(a, b1, acc[ms][1]);
      }
    }
    __syncthreads();                     // reads done before buffer reuse
  }

#pragma unroll
  for (int ns = 0; ns < 2; ++ns) {
    const int colC = nBase + col0 + ns * 16 + lm;
    const float bv = bias[colC];
#pragma unroll
    for (int ms = 0; ms < 4; ++ms) {
#pragma unroll
      for (int r = 0; r < 8; ++r) {
        const int rowC = mBase + row0 + ms * 16 + r + half * 8;
        C[(size_t)rowC * N + colC] = acc[ms][ns][r] + bv;
      }
    }
  }
}

// ===========================================================================
// Kernel 2/3: masked attention, one (b, h) per 128-thread block (4 waves).
// ===========================================================================
#define QVS 132  // q/k/v LDS stride (float4-aligned rows)
#define SS2 34   // score stride (even -> 8B-aligned v2f loads)

__global__ void __launch_bounds__(128)
attn_kernel(const float* __restrict__ qkv,            // (B*N, 1536)
            const unsigned char* __restrict__ pad,    // (B, N) bool
            float* __restrict__ ctx)                  // (B*N, 512)
{
  __shared__ float sQ[NA * QVS], sK[NA * QVS], sV[NA * QVS];
  __shared__ float sS[NA * SS2];

  const int b = blockIdx.x, h = blockIdx.y;
  const int tid  = threadIdx.x;
  const int wave = tid >> 5, lane = tid & 31;
  const int half = lane >> 4, lm = lane & 15;

  const size_t base = (size_t)b * NA * D3 + h * HDIM;
#pragma unroll
  for (int i = 0; i < 8; ++i) {
    int g = tid + i * 128, r = g >> 5, c4 = (g & 31) << 2;
    size_t o = base + (size_t)r * D3 + c4;
    *(float4*)&sQ[r * QVS + c4] = *(const float4*)&qkv[o];
    *(float4*)&sK[r * QVS + c4] = *(const float4*)&qkv[o + DMODEL];
    *(float4*)&sV[r * QVS + c4] = *(const float4*)&qkv[o + 2 * DMODEL];
  }
  __syncthreads();

  {
    const int m0 = (wave >> 1) * 16, n0 = (wave & 1) * 16;
    v8f c = {};
#pragma unroll 4
    for (int kb = 0; kb < HDIM; kb += 4) {
      const int col = kb + half * 2;
      v2f a  = *(const v2f*)&sQ[(m0 + lm) * QVS + col];
      v2f bb = *(const v2f*)&sK[(n0 + lm) * QVS + col];  // B[c][n] = K[n][c]
      c = wmma4(a, bb, c);
    }
#pragma unroll
    for (int r = 0; r < 8; ++r)
      sS[(m0 + r + half * 8) * SS2 + n0 + lm] = c[r] * 0.08838834764831845f;
  }
  __syncthreads();

  if (tid < NA) {
    float* row = &sS[tid * SS2];
    const unsigned char* pd = pad + (size_t)b * NA;
    float mx = -INFINITY;
#pragma unroll
    for (int j = 0; j < NA; ++j)
      if (j != tid && !pd[j]) mx = fmaxf(mx, row[j]);
    float sum = 0.f;
#pragma unroll
    for (int j = 0; j < NA; ++j) {
      float e = (j != tid && !pd[j]) ? __expf(row[j] - mx) : 0.f;
      row[j] = e;
      sum += e;
    }
    const float inv = sum > 0.f ? 1.f / sum : 0.f;
#pragma unroll
    for (int j = 0; j < NA; ++j) row[j] *= inv;
  }
  __syncthreads();

  float* dst = ctx + (size_t)b * NA * DMODEL + h * HDIM;
  for (int id = wave; id < 16; id += 4) {
    const int m0 = (id >> 3) * 16, n0 = (id & 7) * 16;
    v8f c = {};
#pragma unroll
    for (int kb = 0; kb < NA; kb += 4) {
      const int col = kb + half * 2;
      v2f a = *(const v2f*)&sS[(m0 + lm) * SS2 + col];
      v2f bb;
      bb.x = sV[col * QVS + n0 + lm];          // B[k][n] = V[k][n]
      bb.y = sV[(col + 1) * QVS + n0 + lm];
      c = wmma4(a, bb, c);
    }
#pragma unroll
    for (int r = 0; r < 8; ++r)
      dst[(size_t)(m0 + r + half * 8) * DMODEL + n0 + lm] = c[r];
  }
}

// ===========================================================================
// Fallback: fully-fused one-workgroup-per-batch kernel (no workspace).
// ===========================================================================
#define XS   516
#define QS   1540
#define SSF  34

__global__ void __launch_bounds__(256, 1)
caa_fused_kernel(const float* __restrict__ X,
                 const unsigned char* __restrict__ pad,
                 const float* __restrict__ W1, const float* __restrict__ b1,
                 const float* __restrict__ W2, const float* __restrict__ b2,
                 float* __restrict__ out)
{
  extern __shared__ float smem[];
  float* sX   = smem;
  float* sQKV = sX + NA * XS;
  float* sS   = sQKV + NA * QS;

  const int b    = blockIdx.x;
  const int tid  = threadIdx.x;
  const int wave = tid >> 5, lane = tid & 31;
  const int half = lane >> 4, lm = lane & 15;

  {
    const float* src = X + (size_t)b * NA * DMODEL;
    __builtin_prefetch(&W1[tid * 512], 0, 1);
    __builtin_prefetch(&W2[(tid & 255) * 512], 0, 1);
#pragma unroll
    for (int i = 0; i < 16; ++i) {
      int g = tid + i * 256, row = g >> 7, col = (g & 127) << 2;
      *(float4*)&sX[row * XS + col] = *(const float4*)&src[g * 4];
    }
  }
  __syncthreads();

  for (int nt = wave; nt < 96; nt += 8) {
    const int j = nt * 16 + lm;
    v8f c0 = {}, c1 = {};
#pragma unroll 4
    for (int kb = 0; kb < DMODEL; kb += 4) {
      const int col = kb + half * 2;
      v2f bw = *(const v2f*)&W1[(size_t)j * DMODEL + col];
      v2f a0 = *(const v2f*)&sX[lm * XS + col];
      v2f a1 = *(const v2f*)&sX[(16 + lm) * XS + col];
      c0 = wmma4(a0, bw, c0);
      c1 = wmma4(a1, bw, c1);
    }
    const float bias = b1[j];
#pragma unroll
    for (int r = 0; r < 8; ++r) {
      sQKV[(r + half * 8) * QS + j]      = c0[r] + bias;
      sQKV[(16 + r + half * 8) * QS + j] = c1[r] + bias;
    }
  }
  __syncthreads();

  for (int id = wave; id < 16; id += 8) {
    const int h = id >> 2, m0 = ((id >> 1) & 1) * 16, n0 = (id & 1) * 16;
    const int qB = h * HDIM, kB = DMODEL + h * HDIM;
    v8f c = {};
#pragma unroll 4
    for (int kb = 0; kb < HDIM; kb += 4) {
      const int col = kb + half * 2;
      v2f a  = *(const v2f*)&sQKV[(m0 + lm) * QS + qB + col];
      v2f bb = *(const v2f*)&sQKV[(n0 + lm) * QS + kB + col];
      c = wmma4(a, bb, c);
    }
#pragma unroll
    for (int r = 0; r < 8; ++r)
      sS[h * (NA * SSF) + (m0 + r + half * 8) * SSF + n0 + lm] =
          c[r] * 0.08838834764831845f;
  }
  __syncthreads();

  if (tid < NHEAD * NA) {
    const int h = tid >> 5, i = tid & 31;
    float* row = &sS[h * (NA * SSF) + i * SSF];
    const unsigned char* pd = pad + (size_t)b * NA;
    float mx = -INFINITY;
#pragma unroll
    for (int j = 0; j < NA; ++j)
      if (j != i && !pd[j]) mx = fmaxf(mx, row[j]);
    float sum = 0.f;
#pragma unroll
    for (int j = 0; j < NA; ++j) {
      float e = (j != i && !pd[j]) ? __expf(row[j] - mx) : 0.f;
      row[j] = e;
      sum += e;
    }
    const float inv = sum > 0.f ? 1.f / sum : 0.f;
#pragma unroll
    for (int j = 0; j < NA; ++j) row[j] *= inv;
  }
  __syncthreads();

  for (int id = wave; id < 64; id += 8) {
    const int h = id >> 4, m0 = ((id >> 3) & 1) * 16, n0 = (id & 7) * 16;
    const int vB = 2 * DMODEL + h * HDIM + n0;
    v8f c = {};
#pragma unroll
    for (int kb = 0; kb < NA; kb += 4) {
      const int col = kb + half * 2;
      v2f a = *(const v2f*)&sS[h * (NA * SSF) + (m0 + lm) * SSF + col];
      v2f bb;
      bb.x = sQKV[col * QS + vB + lm];
      bb.y = sQKV[(col + 1) * QS + vB + lm];
      c = wmma4(a, bb, c);
    }
#pragma unroll
    for (int r = 0; r < 8; ++r)
      sX[(m0 + r + half * 8) * XS + h * HDIM + n0 + lm] = c[r];
  }
  __syncthreads();

  float* dst = out + (size_t)b * NA * DMODEL;
  for (int nt = wave; nt < 32; nt += 8) {
    const int j = nt * 16 + lm;
    v8f c0 = {}, c1 = {};
#pragma unroll 4
    for (int kb = 0; kb < DMODEL; kb += 4) {
      const int col = kb + half * 2;
      v2f bw = *(const v2f*)&W2[(size_t)j * DMODEL + col];
      v2f a0 = *(const v2f*)&sX[lm * XS + col];
      v2f a1 = *(const v2f*)&sX[(16 + lm) * XS + col];
      c0 = wmma4(a0, bw, c0);
      c1 = wmma4(a1, bw, c1);
    }
    const float bias = b2[j];
#pragma unroll
    for (int r = 0; r < 8; ++r) {
      dst[(r + half * 8) * DMODEL + j]      = c0[r] + bias;
      dst[(16 + r + half * 8) * DMODEL + j] = c1[r] + bias;
    }
  }
}

// ===========================================================================
extern "C" void kernel_launch(void* const* d_in, const int* in_sizes, int n_in,
                              void* d_out, int out_size, void* d_ws, size_t ws_size,
                              hipStream_t stream) {
  const float*         X   = (const float*)d_in[0];
  const unsigned char* pad = (const unsigned char*)d_in[1];
  const float*         W1  = (const float*)d_in[2];
  const float*         b1  = (const float*)d_in[3];
  const float*         W2  = (const float*)d_in[4];
  const float*         b2  = (const float*)d_in[5];
  float*               out = (float*)d_out;

  const size_t QKV_BYTES = (size_t)MTOT * D3 * sizeof(float);     // 192 MB
  const size_t CTX_BYTES = (size_t)MTOT * DMODEL * sizeof(float); //  64 MB

  if (ws_size >= QKV_BYTES + CTX_BYTES) {
    float* qkv = (float*)d_ws;
    float* ctx = (float*)((char*)d_ws + QKV_BYTES);
    gemm_nt_bias_kernel<<<dim3(MTOT / 128, D3 / 128), 256, 0, stream>>>(
        X, W1, b1, qkv, D3);
    attn_kernel<<<dim3(BATCH, NHEAD), 128, 0, stream>>>(qkv, pad, ctx);
    gemm_nt_bias_kernel<<<dim3(MTOT / 128, DMODEL / 128), 256, 0, stream>>>(
        ctx, W2, b2, out, DMODEL);
  } else {
    const size_t smemBytes =
        (size_t)(NA * XS + NA * QS + NHEAD * NA * SSF) * sizeof(float);
    hipFuncSetAttribute((const void*)caa_fused_kernel,
                        hipFuncAttributeMaxDynamicSharedMemorySize,
                        (int)smemBytes);
    caa_fused_kernel<<<BATCH, 256, smemBytes, stream>>>(
        X, pad, W1, b1, W2, b2, out);
  }
}